// CustomGraphConv_37666863186139
// MI455X (gfx1250) — compile-verified
//
#include <hip/hip_runtime.h>

// ---------------------------------------------------------------------------
// CustomGraphConv for MI455X (gfx1250, wave32).
//
//   z[n, a*16+o] = sum_i  W[a,o,i] * x[n,i]          (dense GEMM -> WMMA f32)
//   msg[e, o]    = sum_a  attr[e,a] * z[src_e, a, o] (gather + FMA, L2-resident z)
//   out[d, o]    = relu( bias[o] + sum_{e: dst==d} msg[e,o] )  (f32 HW atomics)
//
// z is [N,128] f32 = 51.2 MB in d_ws: fits in the 192 MB L2, so the random
// src-gather in the edge phase stays on-chip after first touch.
// ---------------------------------------------------------------------------

typedef __attribute__((ext_vector_type(2))) float v2f;
typedef __attribute__((ext_vector_type(8))) float v8f;

#define IN_C    16
#define OUT_C   16
#define N_EATTR 8
#define ZC      (N_EATTR * OUT_C)   // 128 precomputed channels per node

// ---------------------------------------------------------------------------
// Kernel 1: z = x @ Wt via V_WMMA_F32_16X16X4_F32, one wave per 16-node tile.
//
// A (16x4 f32, ISA 7.12.2):  lane = M + 16*(K>=2), vgpr = K&1
//   -> a.x = X[m][4*kc + 2*hi], a.y = X[m][4*kc + 2*hi + 1]
// B (4x16 f32), same reg/lane-half pattern as C/D:
//   -> b.x = Wt[4*kc + 2*hi][n], b.y = Wt[4*kc + 2*hi + 1][n]
//   where Wt[i][c] = weight_matrix flat [c*16 + i],  c = a*16 + n
// D (16x16 f32, 8 VGPRs): element p -> row m = p + 8*hi, col n = lane&15
// ---------------------------------------------------------------------------
__global__ __launch_bounds__(32)
void gconv_precompute_z_wmma(const float* __restrict__ x,   // [N,16]
                             const float* __restrict__ w,   // [8,16,16] flat (a*16+o)*16+i
                             float* __restrict__ z,         // [N,128]
                             int n_nodes)
{
    const int nodeBase = blockIdx.x * 16;
    const int lane = threadIdx.x;       // 0..31, full wave active (EXEC all 1s)
    const int hi   = lane >> 4;
    const int lo   = lane & 15;

    // A fragments for all 4 K-chunks (K=16 total); clamp row for tail tiles
    // (clamping keeps EXEC all-1s, required by WMMA).
    int arow = nodeBase + lo;
    if (arow >= n_nodes) arow = n_nodes - 1;
    const float* xrow = x + (long)arow * IN_C;
    v2f afrag[4];
#pragma unroll
    for (int kc = 0; kc < 4; ++kc) {
        const int k0 = 4 * kc + 2 * hi;
        afrag[kc].x = xrow[k0];
        afrag[kc].y = xrow[k0 + 1];
    }

    // Per-lane output base: element (p, a) lands at byte offset p*512 + a*64,
    // all compile-time immediates -> bare global_store_b32 with IOFFSET.
    float* zb = z + (long)(nodeBase + 8 * hi) * ZC + lo;
    const bool fullTile = (nodeBase + 16 <= n_nodes);

    if (fullTile) {
#pragma unroll
        for (int a = 0; a < N_EATTR; ++a) {
            const float* wc = w + (a * 16 + lo) * IN_C;   // Wt[:, c]
            v8f d = {};
#pragma unroll
            for (int kc = 0; kc < 4; ++kc) {
                const int k0 = 4 * kc + 2 * hi;
                v2f b;
                b.x = wc[k0];
                b.y = wc[k0 + 1];
                d = __builtin_amdgcn_wmma_f32_16x16x4_f32(false, afrag[kc],
                                                          false, b,
                                                          (short)0, d,
                                                          false, false);
            }
#pragma unroll
            for (int p = 0; p < 8; ++p)
                zb[(long)p * ZC + a * 16] = d[p];
        }
    } else {
#pragma unroll
        for (int a = 0; a < N_EATTR; ++a) {
            const float* wc = w + (a * 16 + lo) * IN_C;
            v8f d = {};
#pragma unroll
            for (int kc = 0; kc < 4; ++kc) {
                const int k0 = 4 * kc + 2 * hi;
                v2f b;
                b.x = wc[k0];
                b.y = wc[k0 + 1];
                d = __builtin_amdgcn_wmma_f32_16x16x4_f32(false, afrag[kc],
                                                          false, b,
                                                          (short)0, d,
                                                          false, false);
            }
#pragma unroll
            for (int p = 0; p < 8; ++p) {
                const int row = nodeBase + p + 8 * hi;
                if (row < n_nodes)
                    z[(long)row * ZC + a * 16 + lo] = d[p];
            }
        }
    }
}

// ---------------------------------------------------------------------------
// Kernel 2: zero the accumulator (d_out is poisoned by the harness).
// ---------------------------------------------------------------------------
__global__ __launch_bounds__(256)
void gconv_zero(float4* __restrict__ out, int n4)
{
    const int i = blockIdx.x * 256 + threadIdx.x;
    if (i < n4) out[i] = make_float4(0.f, 0.f, 0.f, 0.f);
}

// ---------------------------------------------------------------------------
// Kernel 3: edge phase. 16 lanes per edge; lane owns out-channel o.
// Lanes 0..7 load the 8 attrs, lanes 8..9 load the int64 src/dst (so the
// 16B of index data is loaded once per group, in parallel with the attrs);
// everything is broadcast with intra-group shuffles. The z gather is a
// coalesced 64B segment per attr channel (L2-resident). The scatter is a
// hardware global_atomic_add_f32.
// ---------------------------------------------------------------------------
__global__ __launch_bounds__(256)
void gconv_edge_scatter(const long long* __restrict__ ei,  // [2,E] int64
                        const float* __restrict__ attr,    // [E,8]
                        const float* __restrict__ z,       // [N,128]
                        float* __restrict__ out,           // [N,16] accumulator
                        int n_edges)
{
    const int t = blockIdx.x * 256 + threadIdx.x;
    const int e = t >> 4;
    if (e >= n_edges) return;          // uniform per 16-lane group

    const int lane = threadIdx.x & 31;
    const int o    = lane & 15;
    const int eg   = lane & 16;        // group base within the wave

    // cooperative loads: lanes 0..7 -> attrs, lanes 8..9 -> src/dst
    float av = 0.0f;
    int   iv = 0;
    if (o < N_EATTR)
        av = attr[(long)e * N_EATTR + o];
    else if (o < N_EATTR + 2)
        iv = (int)ei[(long)(o - N_EATTR) * n_edges + e];

    const int src = __shfl(iv, eg | 8, 32);
    const int dst = __shfl(iv, eg | 9, 32);

    const float* zp = z + (long)src * ZC + o;
    float acc = 0.0f;
#pragma unroll
    for (int a = 0; a < N_EATTR; ++a) {
        const float aa = __shfl(av, eg | a, 32);
        acc += aa * zp[a * 16];
    }

    unsafeAtomicAdd(&out[(long)dst * OUT_C + o], acc);  // global_atomic_add_f32
}

// ---------------------------------------------------------------------------
// Kernel 4: out = relu(out + bias), in place, float4-vectorized.
// i&3 selects which quarter of the 16-channel bias row this float4 covers.
// ---------------------------------------------------------------------------
__global__ __launch_bounds__(256)
void gconv_bias_relu(float4* __restrict__ out, const float4* __restrict__ bias,
                     int n4)
{
    const int i = blockIdx.x * 256 + threadIdx.x;
    if (i < n4) {
        const float4 b = bias[i & 3];
        float4 v = out[i];
        v.x = fmaxf(v.x + b.x, 0.f);
        v.y = fmaxf(v.y + b.y, 0.f);
        v.z = fmaxf(v.z + b.z, 0.f);
        v.w = fmaxf(v.w + b.w, 0.f);
        out[i] = v;
    }
}

// ---------------------------------------------------------------------------
extern "C" void kernel_launch(void* const* d_in, const int* in_sizes, int n_in,
                              void* d_out, int out_size, void* d_ws, size_t ws_size,
                              hipStream_t stream)
{
    const float*     x    = (const float*)d_in[0];
    const long long* ei   = (const long long*)d_in[1];   // int64 [2,E]
    const float*     attr = (const float*)d_in[2];       // [E,8]
    const float*     w    = (const float*)d_in[3];       // [8,16,16]
    const float*     bias = (const float*)d_in[4];       // [16]
    float*           out  = (float*)d_out;               // [N,16]

    const int N = in_sizes[0] / IN_C;
    const int E = in_sizes[2] / N_EATTR;

    float* z = (float*)d_ws;                             // [N,128] = 51.2 MB

    const int nTiles = (N + 15) / 16;
    gconv_precompute_z_wmma<<<nTiles, 32, 0, stream>>>(x, w, z, N);

    const int n4 = (N * OUT_C) / 4;                      // OUT_C==16 -> exact
    gconv_zero<<<(n4 + 255) / 256, 256, 0, stream>>>((float4*)out, n4);

    const long edgeThreads = (long)E * 16;
    gconv_edge_scatter<<<(int)((edgeThreads + 255) / 256), 256, 0, stream>>>(
        ei, attr, z, out, E);

    gconv_bias_relu<<<(n4 + 255) / 256, 256, 0, stream>>>(
        (float4*)out, (const float4*)bias, n4);
}